// GraphSAGE_gc_1219770712268
// MI455X (gfx1250) — compile-verified
//
#include <hip/hip_runtime.h>
#include <hip/hip_bf16.h>

typedef float v2f __attribute__((ext_vector_type(2)));
typedef float v8f __attribute__((ext_vector_type(8)));

#define DFEAT 64

// ---------------------------------------------------------------- utilities
__global__ void zero_f32_kernel(float* __restrict__ p, int n) {
    int t = blockIdx.x * blockDim.x + threadIdx.x;
    if (t < n) p[t] = 0.0f;
}

// in-degree count: cnt[dst[e]] += 1
__global__ void degree_kernel(const int* __restrict__ dst, float* __restrict__ cnt, int E) {
    int e = blockIdx.x * blockDim.x + threadIdx.x;
    if (e < E) atomicAdd(&cnt[dst[e]], 1.0f);
}

// in-place cnt -> 1 / max(cnt, 1)  (hoists the f32 division out of hot kernels)
__global__ void invert_kernel(float* __restrict__ cnt, int N) {
    int n = blockIdx.x * blockDim.x + threadIdx.x;
    if (n < N) cnt[n] = 1.0f / fmaxf(cnt[n], 1.0f);
}

// per-graph node count
__global__ void graph_count_kernel(const int* __restrict__ batch, float* __restrict__ gcnt, int N) {
    int n = blockIdx.x * blockDim.x + threadIdx.x;
    if (n < N) atomicAdd(&gcnt[batch[n]], 1.0f);
}

// ------------------------------------------------------- edge gather/scatter
// 16 lanes per edge; each lane moves a float4 slice of the 64-float feature row.
__global__ void scatter_kernel(const float* __restrict__ h,
                               const int* __restrict__ src,
                               const int* __restrict__ dst,
                               float* __restrict__ aggr, int E) {
    long t = (long)blockIdx.x * blockDim.x + threadIdx.x;
    long e = t >> 4;
    if (e >= E) return;
    int f = (int)(t & 15) << 2;
    int s = src[e];
    int d = dst[e];
    const float4 v = *(const float4*)(h + (long)s * DFEAT + f);
    float* p = aggr + (long)d * DFEAT + f;
    atomicAdd(p + 0, v.x);
    atomicAdd(p + 1, v.y);
    atomicAdd(p + 2, v.z);
    atomicAdd(p + 3, v.w);
}

// ------------------------------------------------- fused SAGE layer via WMMA
// out = relu( (aggr*invdeg) @ Wl^T  +  h @ Wr^T  +  b )
// Each wave computes a 16(M) x 64(N) strip with FOUR independent
// V_WMMA_F32_16X16X4_F32 accumulators (breaks the C->D RAW chain and reads
// each A fragment once). Block = 256 threads = 8 waves -> 128 rows per block.
__global__ void sage_wmma_kernel(const float* __restrict__ aggr,
                                 const float* __restrict__ h,
                                 const float* __restrict__ invdeg,
                                 const float* __restrict__ Wl,   // [64][64] row-major
                                 const float* __restrict__ Wr,   // [64][64]
                                 const float* __restrict__ bias, // [64]
                                 float* __restrict__ out,
                                 int do_relu) {
    const int lane = threadIdx.x & 31;
    const int wave = threadIdx.x >> 5;
    const int r    = lane & 15;
    const int half = lane >> 4;           // 0: K pair {0,1}; 1: K pair {2,3}
    const int m0   = blockIdx.x * 128 + wave * 16;

    const int m = m0 + r;                 // A-matrix row this lane owns
    const float inv = invdeg[m];
    const float* __restrict__ arow = aggr + (long)m * DFEAT;
    const float* __restrict__ hrow = h    + (long)m * DFEAT;
    // B[k][n] = W[n][k]; lane supplies column n = n0 + r for each 16-wide n-tile
    const float* __restrict__ wl0 = Wl + (0  + r) * DFEAT;
    const float* __restrict__ wl1 = Wl + (16 + r) * DFEAT;
    const float* __restrict__ wl2 = Wl + (32 + r) * DFEAT;
    const float* __restrict__ wl3 = Wl + (48 + r) * DFEAT;
    const float* __restrict__ wr0 = Wr + (0  + r) * DFEAT;
    const float* __restrict__ wr1 = Wr + (16 + r) * DFEAT;
    const float* __restrict__ wr2 = Wr + (32 + r) * DFEAT;
    const float* __restrict__ wr3 = Wr + (48 + r) * DFEAT;

    v8f acc0 = {}, acc1 = {}, acc2 = {}, acc3 = {};

    // (aggr * invdeg) @ Wl^T : K = 64 in 16 steps of 4, 4 n-tiles per step
#pragma unroll
    for (int kk = 0; kk < 16; ++kk) {
        const int k = kk * 4 + half * 2;
        v2f a = *(const v2f*)(arow + k);
        a.x *= inv; a.y *= inv;
        const v2f b0 = *(const v2f*)(wl0 + k);
        const v2f b1 = *(const v2f*)(wl1 + k);
        const v2f b2 = *(const v2f*)(wl2 + k);
        const v2f b3 = *(const v2f*)(wl3 + k);
        acc0 = __builtin_amdgcn_wmma_f32_16x16x4_f32(false, a, false, b0, (short)0, acc0, false, false);
        acc1 = __builtin_amdgcn_wmma_f32_16x16x4_f32(false, a, false, b1, (short)0, acc1, false, false);
        acc2 = __builtin_amdgcn_wmma_f32_16x16x4_f32(false, a, false, b2, (short)0, acc2, false, false);
        acc3 = __builtin_amdgcn_wmma_f32_16x16x4_f32(false, a, false, b3, (short)0, acc3, false, false);
    }
    // + h @ Wr^T
#pragma unroll
    for (int kk = 0; kk < 16; ++kk) {
        const int k = kk * 4 + half * 2;
        const v2f a  = *(const v2f*)(hrow + k);
        const v2f b0 = *(const v2f*)(wr0 + k);
        const v2f b1 = *(const v2f*)(wr1 + k);
        const v2f b2 = *(const v2f*)(wr2 + k);
        const v2f b3 = *(const v2f*)(wr3 + k);
        acc0 = __builtin_amdgcn_wmma_f32_16x16x4_f32(false, a, false, b0, (short)0, acc0, false, false);
        acc1 = __builtin_amdgcn_wmma_f32_16x16x4_f32(false, a, false, b1, (short)0, acc1, false, false);
        acc2 = __builtin_amdgcn_wmma_f32_16x16x4_f32(false, a, false, b2, (short)0, acc2, false, false);
        acc3 = __builtin_amdgcn_wmma_f32_16x16x4_f32(false, a, false, b3, (short)0, acc3, false, false);
    }

    // epilogue: +bias, optional relu, store.
    // C/D layout: vgpr i -> row m0 + 8*half + i, col n0 + (lane & 15)
    const float bv0 = bias[0  + r];
    const float bv1 = bias[16 + r];
    const float bv2 = bias[32 + r];
    const float bv3 = bias[48 + r];
    const int row_base = m0 + half * 8;
#pragma unroll
    for (int i = 0; i < 8; ++i) {
        float* orow = out + (long)(row_base + i) * DFEAT + r;
        float v0 = acc0[i] + bv0;
        float v1 = acc1[i] + bv1;
        float v2 = acc2[i] + bv2;
        float v3 = acc3[i] + bv3;
        if (do_relu) {
            v0 = fmaxf(v0, 0.0f); v1 = fmaxf(v1, 0.0f);
            v2 = fmaxf(v2, 0.0f); v3 = fmaxf(v3, 0.0f);
        }
        orow[0]  = v0;
        orow[16] = v1;
        orow[32] = v2;
        orow[48] = v3;
    }
}

// --------------------------------- output layer (64 -> 2) fused with pooling
__global__ void sage_out_pool_kernel(const float* __restrict__ aggr,
                                     const float* __restrict__ h,
                                     const float* __restrict__ invdeg,
                                     const float* __restrict__ Wlo, // [2][64]
                                     const float* __restrict__ Wro, // [2][64]
                                     const float* __restrict__ bo,  // [2]
                                     const int* __restrict__ batch,
                                     float* __restrict__ gsum, int N) {
    int n = blockIdx.x * blockDim.x + threadIdx.x;
    if (n >= N) return;
    const float inv = invdeg[n];
    const float* ar = aggr + (long)n * DFEAT;
    const float* hr = h    + (long)n * DFEAT;
    float acc0 = bo[0], acc1 = bo[1];
#pragma unroll 8
    for (int k = 0; k < DFEAT; ++k) {
        const float av = ar[k] * inv;
        const float hv = hr[k];
        acc0 += av * Wlo[k]         + hv * Wro[k];
        acc1 += av * Wlo[DFEAT + k] + hv * Wro[DFEAT + k];
    }
    const int g = batch[n];
    atomicAdd(&gsum[g * 2 + 0], acc0);
    atomicAdd(&gsum[g * 2 + 1], acc1);
}

__global__ void finalize_kernel(const float* __restrict__ gsum,
                                const float* __restrict__ gcnt,
                                float* __restrict__ out) {
    int t = threadIdx.x;                  // 256 = 128 graphs x 2 classes
    out[t] = gsum[t] / fmaxf(gcnt[t >> 1], 1.0f);
}

// ---------------------------------------------------------------------------
extern "C" void kernel_launch(void* const* d_in, const int* in_sizes, int n_in,
                              void* d_out, int out_size, void* d_ws, size_t ws_size,
                              hipStream_t stream) {
    const float* x     = (const float*)d_in[0];
    const int*   ei    = (const int*)  d_in[1];
    const int*   batch = (const int*)  d_in[2];
    const float* Wl    = (const float*)d_in[3];  // [3][64][64]
    const float* Wr    = (const float*)d_in[4];
    const float* b     = (const float*)d_in[5];  // [3][64]
    const float* Wlo   = (const float*)d_in[6];  // [2][64]
    const float* Wro   = (const float*)d_in[7];
    const float* bo    = (const float*)d_in[8];  // [2]

    const int N = in_sizes[0] / DFEAT;           // 64000
    const int E = in_sizes[1] / 2;               // 1024000
    const int* src = ei;
    const int* dst = ei + E;

    const size_t feat = (size_t)N * DFEAT;       // 4,096,000 floats
    float* ws   = (float*)d_ws;
    float* aggr = ws;                            // feat
    float* hA   = aggr + feat;                   // feat
    float* hB   = hA   + feat;                   // feat
    float* cnt  = hB   + feat;                   // N   (becomes 1/max(deg,1))
    float* gsum = cnt  + N;                      // 256
    float* gcnt = gsum + 256;                    // 128

    const int B256 = 256;
    const int featBlocks = (int)((feat + B256 - 1) / B256);
    const int scatBlocks = (int)(((long)E * 16 + B256 - 1) / B256);
    const int nBlocks    = (N + B256 - 1) / B256;

    // per-launch init (deterministic: same work every call)
    zero_f32_kernel<<<nBlocks, B256, 0, stream>>>(cnt, N);
    zero_f32_kernel<<<1, B256, 0, stream>>>(gsum, 256);
    zero_f32_kernel<<<1, 128, 0, stream>>>(gcnt, 128);
    degree_kernel<<<(E + B256 - 1) / B256, B256, 0, stream>>>(dst, cnt, E);
    graph_count_kernel<<<nBlocks, B256, 0, stream>>>(batch, gcnt, N);
    invert_kernel<<<nBlocks, B256, 0, stream>>>(cnt, N);

    const float* hin = x;
    float* bufs[2] = {hA, hB};
    for (int l = 0; l < 3; ++l) {
        zero_f32_kernel<<<featBlocks, B256, 0, stream>>>(aggr, (int)feat);
        scatter_kernel<<<scatBlocks, B256, 0, stream>>>(hin, src, dst, aggr, E);
        sage_wmma_kernel<<<N / 128, B256, 0, stream>>>(
            aggr, hin, cnt, Wl + l * DFEAT * DFEAT, Wr + l * DFEAT * DFEAT,
            b + l * DFEAT, bufs[l & 1], /*do_relu=*/1);
        hin = bufs[l & 1];
    }

    // output SAGEConv (64 -> 2) + global mean pool
    zero_f32_kernel<<<featBlocks, B256, 0, stream>>>(aggr, (int)feat);
    scatter_kernel<<<scatBlocks, B256, 0, stream>>>(hin, src, dst, aggr, E);
    sage_out_pool_kernel<<<nBlocks, B256, 0, stream>>>(
        aggr, hin, cnt, Wlo, Wro, bo, batch, gsum, N);
    finalize_kernel<<<1, 256, 0, stream>>>(gsum, gcnt, (float*)d_out);
}